// AutoHybridModel_84250078478771
// MI455X (gfx1250) — compile-verified
//
#include <hip/hip_runtime.h>
#include <hip/hip_bf16.h>

#define BSZ  8192
#define LSEQ 100
#define NFEAT 384   // 4 fields * 6 pools * 16 dims

typedef float v2f __attribute__((ext_vector_type(2)));
typedef float v8f __attribute__((ext_vector_type(8)));

// ---------------------------------------------------------------------------
// Phase 1: per-(row, field) pooling. pooled layout: [feature][b], feature =
// field*96 + pool*16 + dim, pool order: sum, mean, max, min, korder, atten.
// ---------------------------------------------------------------------------
__global__ __launch_bounds__(256) void pool_kernel(
    const int* __restrict__ ids_ug, const int* __restrict__ ids_urb,
    const int* __restrict__ ids_mg, const int* __restrict__ ids_mt,
    const int* __restrict__ len_ug, const int* __restrict__ len_urb,
    const int* __restrict__ len_mg, const int* __restrict__ len_mt,
    const float* __restrict__ emb_movie, const float* __restrict__ emb_tag,
    const float* __restrict__ emb_genre,
    const float* __restrict__ att_movie, const float* __restrict__ att_tag,
    const float* __restrict__ att_genre,
    float* __restrict__ pooled)
{
    int tid = blockIdx.x * blockDim.x + threadIdx.x;
    if (tid >= 4 * BSZ) return;
    int field = tid >> 13;        // BSZ == 2^13
    int b     = tid & (BSZ - 1);

    const int*   ids;  const int* lenp;
    const float* tab;  const float* atab;
    switch (field) {
      case 0:  ids = ids_ug;  lenp = len_ug;  tab = emb_genre; atab = att_genre; break;
      case 1:  ids = ids_urb; lenp = len_urb; tab = emb_movie; atab = att_movie; break;
      case 2:  ids = ids_mg;  lenp = len_mg;  tab = emb_genre; atab = att_genre; break;
      default: ids = ids_mt;  lenp = len_mt;  tab = emb_tag;   atab = att_tag;   break;
    }
    int len = lenp[b];

    float s[16], q[16], at[16];
    #pragma unroll
    for (int d = 0; d < 16; ++d) { s[d] = 0.f; q[d] = 0.f; at[d] = 0.f; }
    float denom = 0.f;
    float maxv = -1.f;  int mxi = 0;      // argmax l2, first occurrence
    float minv = 1e30f; int mni = 0;      // argmin of (l2==0 ? 9999 : l2)

    for (int l = 0; l < LSEQ; ++l) {
        int id = ids[b * LSEQ + l];
        float m = (l < len) ? 1.f : 0.f;
        const float4* t = (const float4*)(tab + (size_t)id * 16);
        float me[16];
        *((float4*)&me[0])  = t[0];
        *((float4*)&me[4])  = t[1];
        *((float4*)&me[8])  = t[2];
        *((float4*)&me[12]) = t[3];
        float aw = atab[id] * m;
        float l2 = 0.f;
        #pragma unroll
        for (int d = 0; d < 16; ++d) {
            float v = me[d] * m; me[d] = v;
            s[d] += v;
            float sq = v * v;
            q[d] += sq;
            l2   += sq;
        }
        float ew = __expf(aw);          // safe range (aw in ~[0, 1.01])
        denom += ew;
        #pragma unroll
        for (int d = 0; d < 16; ++d) at[d] += me[d] * ew;
        if (l2 > maxv) { maxv = l2; mxi = l; }
        float lv = (l2 == 0.f) ? 9999.f : l2;
        if (lv < minv) { minv = lv; mni = l; }
    }

    // re-gather max/min rows
    float mx[16], mn[16];
    {
        int idm = ids[b * LSEQ + mxi];
        float mm = (mxi < len) ? 1.f : 0.f;
        const float4* t = (const float4*)(tab + (size_t)idm * 16);
        *((float4*)&mx[0]) = t[0]; *((float4*)&mx[4]) = t[1];
        *((float4*)&mx[8]) = t[2]; *((float4*)&mx[12]) = t[3];
        #pragma unroll
        for (int d = 0; d < 16; ++d) mx[d] *= mm;
    }
    {
        int idn = ids[b * LSEQ + mni];
        float mm = (mni < len) ? 1.f : 0.f;
        const float4* t = (const float4*)(tab + (size_t)idn * 16);
        *((float4*)&mn[0]) = t[0]; *((float4*)&mn[4]) = t[1];
        *((float4*)&mn[8]) = t[2]; *((float4*)&mn[12]) = t[3];
        #pragma unroll
        for (int d = 0; d < 16; ++d) mn[d] *= mm;
    }

    // korder
    float ko[16], kn = 0.f;
    #pragma unroll
    for (int d = 0; d < 16; ++d) {
        ko[d] = 0.5f * (s[d] * s[d] - q[d]);
        kn += ko[d] * ko[d];
    }
    float kinv = 1.f / fmaxf(sqrtf(kn), 1e-12f);
    float rden = 1.f / denom;

    size_t base = (size_t)field * 96;
    #pragma unroll
    for (int d = 0; d < 16; ++d) {
        pooled[(base +  0 + d) * BSZ + b] = s[d];            // sum
        pooled[(base + 16 + d) * BSZ + b] = s[d] * 0.01f;    // mean over full L
        pooled[(base + 32 + d) * BSZ + b] = mx[d];           // max
        pooled[(base + 48 + d) * BSZ + b] = mn[d];           // min
        pooled[(base + 64 + d) * BSZ + b] = ko[d] * kinv;    // korder (normalized)
        pooled[(base + 80 + d) * BSZ + b] = at[d] * rden;    // attention
    }
}

// ---------------------------------------------------------------------------
// Phase 2: batch statistics per feature column (biased variance, BN eps 1e-5).
// ---------------------------------------------------------------------------
__global__ __launch_bounds__(256) void stats_kernel(
    const float* __restrict__ pooled, float* __restrict__ mu, float* __restrict__ rs)
{
    int f = blockIdx.x;
    const float* col = pooled + (size_t)f * BSZ;
    float sum = 0.f, sq = 0.f;
    for (int b = threadIdx.x; b < BSZ; b += 256) {
        float v = col[b];
        sum += v; sq += v * v;
    }
    __shared__ float ssum[256], ssq[256];
    ssum[threadIdx.x] = sum; ssq[threadIdx.x] = sq;
    __syncthreads();
    for (int off = 128; off > 0; off >>= 1) {
        if (threadIdx.x < off) {
            ssum[threadIdx.x] += ssum[threadIdx.x + off];
            ssq[threadIdx.x]  += ssq[threadIdx.x + off];
        }
        __syncthreads();
    }
    if (threadIdx.x == 0) {
        float m   = ssum[0] * (1.f / BSZ);
        float var = ssq[0] * (1.f / BSZ) - m * m;
        mu[f] = m;
        rs[f] = rsqrtf(var + 1e-5f);
    }
}

// ---------------------------------------------------------------------------
// Phase 3: BN apply + feature assembly + WMMA MLP (f32 16x16x4) + sigmoid.
// Block = 128 threads (4 waves); each wave owns 2 groups of 16 rows.
// ---------------------------------------------------------------------------
__global__ __launch_bounds__(128) void mlp_kernel(
    const int* __restrict__ uid, const int* __restrict__ mid, const int* __restrict__ yr,
    const float* __restrict__ emb_user, const float* __restrict__ emb_movie,
    const float* __restrict__ emb_year,
    const float* __restrict__ pooled, const float* __restrict__ mu, const float* __restrict__ rs,
    const float* __restrict__ gamma, const float* __restrict__ beta,
    const float* __restrict__ alpha,
    const float* __restrict__ W1, const float* __restrict__ b1,
    const float* __restrict__ W2, const float* __restrict__ b2,
    const float* __restrict__ W3, const float* __restrict__ b3,
    float* __restrict__ out)
{
    __shared__ float xbuf[128 * 112];     // 57,344 B static LDS
    int tid = threadIdx.x;
    int b   = blockIdx.x * 128 + tid;
    float* xr = &xbuf[tid * 112];

    // --- part (a): assemble x[112] for this row into LDS ---
    {
        float4* xv = (float4*)xr;
        const float4* tu = (const float4*)(emb_user  + (size_t)uid[b] * 16);
        xv[0] = tu[0]; xv[1] = tu[1]; xv[2]  = tu[2]; xv[3]  = tu[3];
        const float4* tm = (const float4*)(emb_movie + (size_t)mid[b] * 16);
        xv[4] = tm[0]; xv[5] = tm[1]; xv[6]  = tm[2]; xv[7]  = tm[3];
        const float4* ty = (const float4*)(emb_year  + (size_t)yr[b]  * 16);
        xv[8] = ty[0]; xv[9] = ty[1]; xv[10] = ty[2]; xv[11] = ty[3];
    }
    for (int i = 0; i < 4; ++i) {
        float w[6]; float wden = 0.f;
        #pragma unroll
        for (int p = 0; p < 6; ++p) { w[p] = __expf(alpha[i * 6 + p]); wden += w[p]; }
        float rw = 1.f / wden;
        for (int d = 0; d < 16; ++d) {
            float acc = 0.f;
            #pragma unroll
            for (int p = 0; p < 6; ++p) {
                int f   = i * 96 + p * 16 + d;
                float v = pooled[(size_t)f * BSZ + b];
                float g = gamma[(i * 6 + p) * 16 + d];
                float bb = beta[(i * 6 + p) * 16 + d];
                acc += (w[p] * rw) * (g * (v - mu[f]) * rs[f] + bb);
            }
            xr[48 + i * 16 + d] = acc;
        }
    }
    __syncthreads();

    // --- part (b): MLP with V_WMMA_F32_16X16X4_F32 ---
    int lane = tid & 31;
    int wv   = tid >> 5;
    int half = lane >> 4;
    int mrow = lane & 15;
    const v8f vzero = {0.f, 0.f, 0.f, 0.f, 0.f, 0.f, 0.f, 0.f};

    for (int gsub = 0; gsub < 2; ++gsub) {
        int g = wv * 2 + gsub;
        float* X = &xbuf[g * 16 * 112];   // 16 rows x 112 floats

        // layer 1: [16x112] @ [112x64] in 4 N-tiles, 28 K-steps
        v8f acc[4];
        #pragma unroll
        for (int n = 0; n < 4; ++n) acc[n] = vzero;
        for (int k = 0; k < 112; k += 4) {
            v2f a;
            a.x = X[mrow * 112 + k + 2 * half + 0];
            a.y = X[mrow * 112 + k + 2 * half + 1];
            #pragma unroll
            for (int n = 0; n < 4; ++n) {
                v2f bf;
                bf.x = W1[(k + 2 * half + 0) * 64 + n * 16 + mrow];
                bf.y = W1[(k + 2 * half + 1) * 64 + n * 16 + mrow];
                acc[n] = __builtin_amdgcn_wmma_f32_16x16x4_f32(
                    false, a, false, bf, (short)0, acc[n], false, false);
            }
        }
        // bias + relu, overlay h1 (16x64) on this group's consumed x region
        float* H1 = X;
        #pragma unroll
        for (int n = 0; n < 4; ++n) {
            float bias = b1[n * 16 + mrow];
            #pragma unroll
            for (int v = 0; v < 8; ++v) {
                float hv = fmaxf(acc[n][v] + bias, 0.f);
                H1[(half * 8 + v) * 64 + n * 16 + mrow] = hv;
            }
        }

        // layer 2: [16x64] @ [64x32] in 2 N-tiles, 16 K-steps
        v8f dacc[2];
        dacc[0] = vzero; dacc[1] = vzero;
        for (int k = 0; k < 64; k += 4) {
            v2f a;
            a.x = H1[mrow * 64 + k + 2 * half + 0];
            a.y = H1[mrow * 64 + k + 2 * half + 1];
            #pragma unroll
            for (int n = 0; n < 2; ++n) {
                v2f bf;
                bf.x = W2[(k + 2 * half + 0) * 32 + n * 16 + mrow];
                bf.y = W2[(k + 2 * half + 1) * 32 + n * 16 + mrow];
                dacc[n] = __builtin_amdgcn_wmma_f32_16x16x4_f32(
                    false, a, false, bf, (short)0, dacc[n], false, false);
            }
        }
        float* H2 = X + 16 * 64;          // h2 (16x32) after h1 region
        #pragma unroll
        for (int n = 0; n < 2; ++n) {
            float bias = b2[n * 16 + mrow];
            #pragma unroll
            for (int v = 0; v < 8; ++v) {
                float hv = fmaxf(dacc[n][v] + bias, 0.f);
                H2[(half * 8 + v) * 32 + n * 16 + mrow] = hv;
            }
        }

        // layer 3: [16x32] @ [32x1] + sigmoid (lanes 0..15, one row each)
        if (lane < 16) {
            float z = b3[0];
            #pragma unroll
            for (int k = 0; k < 32; ++k) z += H2[lane * 32 + k] * W3[k];
            out[blockIdx.x * 128 + g * 16 + lane] = 1.f / (1.f + __expf(-z));
        }
    }
}

// ---------------------------------------------------------------------------
extern "C" void kernel_launch(void* const* d_in, const int* in_sizes, int n_in,
                              void* d_out, int out_size, void* d_ws, size_t ws_size,
                              hipStream_t stream) {
    const int* uid     = (const int*)d_in[0];
    const int* mid     = (const int*)d_in[1];
    const int* yr      = (const int*)d_in[2];
    const int* ids_ug  = (const int*)d_in[3];
    const int* ids_urb = (const int*)d_in[4];
    const int* ids_mg  = (const int*)d_in[5];
    const int* ids_mt  = (const int*)d_in[6];
    const int* len_ug  = (const int*)d_in[7];
    const int* len_urb = (const int*)d_in[8];
    const int* len_mg  = (const int*)d_in[9];
    const int* len_mt  = (const int*)d_in[10];
    const float* emb_user  = (const float*)d_in[11];
    const float* emb_movie = (const float*)d_in[12];
    const float* emb_tag   = (const float*)d_in[13];
    const float* emb_genre = (const float*)d_in[14];
    const float* emb_year  = (const float*)d_in[15];
    const float* att_movie = (const float*)d_in[16];
    const float* att_tag   = (const float*)d_in[17];
    const float* att_genre = (const float*)d_in[18];
    const float* gamma = (const float*)d_in[19];
    const float* beta  = (const float*)d_in[20];
    const float* alpha = (const float*)d_in[21];
    const float* W1 = (const float*)d_in[22]; const float* b1 = (const float*)d_in[23];
    const float* W2 = (const float*)d_in[24]; const float* b2 = (const float*)d_in[25];
    const float* W3 = (const float*)d_in[26]; const float* b3 = (const float*)d_in[27];
    float* out = (float*)d_out;

    float* pooled = (float*)d_ws;                       // [384][8192]
    float* mu     = pooled + (size_t)NFEAT * BSZ;       // [384]
    float* rs     = mu + NFEAT;                         // [384]

    pool_kernel<<<(4 * BSZ) / 256, 256, 0, stream>>>(
        ids_ug, ids_urb, ids_mg, ids_mt,
        len_ug, len_urb, len_mg, len_mt,
        emb_movie, emb_tag, emb_genre,
        att_movie, att_tag, att_genre, pooled);

    stats_kernel<<<NFEAT, 256, 0, stream>>>(pooled, mu, rs);

    mlp_kernel<<<BSZ / 128, 128, 0, stream>>>(
        uid, mid, yr, emb_user, emb_movie, emb_year,
        pooled, mu, rs, gamma, beta, alpha,
        W1, b1, W2, b2, W3, b3, out);
}